// CLAGL_79774722556251
// MI455X (gfx1250) — compile-verified
//
#include <hip/hip_runtime.h>
#include <stdint.h>

#ifndef __has_builtin
#define __has_builtin(x) 0
#endif

#if __has_builtin(__builtin_amdgcn_global_load_async_to_lds_b64) && \
    __has_builtin(__builtin_amdgcn_s_wait_asynccnt)
#define CLAGL_ASYNC 1
#else
#define CLAGL_ASYNC 0
#endif

#define CLAGL_BLOCK 256
#define CLAGL_WAVES (CLAGL_BLOCK / 32)
#define CLAGL_MAXBLK 4096   // persistent-ish grid: waves iterate several batches

#if CLAGL_ASYNC
// Builtin signature (from hipcc diagnostic): params are pointers to
// int __attribute__((vector_size(8))) in AS1 (global) / AS3 (LDS).
typedef int clagl_v2i __attribute__((vector_size(8)));
typedef __attribute__((address_space(1))) clagl_v2i* clagl_g64p;
typedef __attribute__((address_space(3))) clagl_v2i* clagl_l64p;
#endif

// ---------------------------------------------------------------------------
// Zero-fill the output (d_out is poisoned before timing).
// ---------------------------------------------------------------------------
__global__ void clagl_zero_f4(float4* __restrict__ p, int n4) {
  int i = blockIdx.x * blockDim.x + threadIdx.x;
  if (i < n4) p[i] = make_float4(0.f, 0.f, 0.f, 0.f);
}

// ---------------------------------------------------------------------------
// COO SpMM scatter: out[row*64 + k] += val * x[col*64 + k]
// Gather source is a virtual concat: col < split -> xa[col], else xb[col-split].
// One wave handles 32 edges per batch; lanes load indices coalesced, each edge
// is broadcast via __shfl; the wave gathers 64 floats (8B/lane, coalesced
// 256B) and scatters with non-returning f32 atomics at agent scope (the whole
// working set is L2-resident on MI455X's 192MB L2).
// Gather path: gfx1250 async-to-LDS loads (ASYNCcnt), 4-deep pipeline.
// ---------------------------------------------------------------------------
__global__ __launch_bounds__(CLAGL_BLOCK) void clagl_spmm_scatter(
    const int* __restrict__ rows, const int* __restrict__ cols,
    const float* __restrict__ vals,
    const float* __restrict__ xa, const float* __restrict__ xb,
    int split, float* __restrict__ out, int nnz) {
#if CLAGL_ASYNC
  __shared__ float lbuf[CLAGL_WAVES][4][64];
#endif
  const int lane  = threadIdx.x & 31;
  const int wid   = threadIdx.x >> 5;
  const int gwave = (blockIdx.x * CLAGL_BLOCK + threadIdx.x) >> 5;
  const int nwave = (gridDim.x * CLAGL_BLOCK) >> 5;

  for (int base = gwave * 32; base < nnz; base += nwave * 32) {
    const int e = base + lane;
    int r = 0, c = 0;
    float v = 0.f;
    if (e < nnz) {
      r = rows[e];
      c = cols[e];
      v = vals[e];
    }
    const int cnt = min(32, nnz - base);

    // Prefetch next batch's edge indices (global_prefetch_b8); live because
    // the grid is capped and each wave iterates multiple batches.
    const int nb = base + nwave * 32;
    if (nb + lane < nnz) {
      __builtin_prefetch(&rows[nb + lane], 0, 3);
      __builtin_prefetch(&cols[nb + lane], 0, 3);
      __builtin_prefetch(&vals[nb + lane], 0, 3);
    }

    auto scatter = [&](int j, float x0, float x1) {
      const float vj = __shfl(v, j);
      const int   rj = __shfl(r, j);
      float* dst = out + (size_t)rj * 64 + 2 * lane;
      __hip_atomic_fetch_add(dst + 0, vj * x0, __ATOMIC_RELAXED,
                             __HIP_MEMORY_SCOPE_AGENT);
      __hip_atomic_fetch_add(dst + 1, vj * x1, __ATOMIC_RELAXED,
                             __HIP_MEMORY_SCOPE_AGENT);
    };

#if CLAGL_ASYNC
    auto issue = [&](int j) {
      const int cj = __shfl(c, j);
      const float* src = (cj < split) ? (xa + (size_t)cj * 64)
                                      : (xb + (size_t)(cj - split) * 64);
      // AS1 global pointer: flat global addresses are bit-identical to AS1.
      // AS3 LDS pointer: proper addrspacecast from the shared-memory pointer.
      __builtin_amdgcn_global_load_async_to_lds_b64(
          (clagl_g64p)(uintptr_t)(src + 2 * lane),
          (clagl_l64p)(void*)&lbuf[wid][j & 3][2 * lane],
          0, 0);
    };
    auto consume = [&](int j) {
      const float2 xv = *(const float2*)&lbuf[wid][j & 3][2 * lane];
      scatter(j, xv.x, xv.y);
    };

    // Prologue: put up to 3 gathers (768B/wave) in flight.
    if (0 < cnt) issue(0);
    if (1 < cnt) issue(1);
    if (2 < cnt) issue(2);

    // Steady state: 4 gathers in flight; wait for the oldest, consume it.
    int j = 0;
    for (; j + 3 < cnt; ++j) {
      issue(j + 3);
      __builtin_amdgcn_s_wait_asynccnt(3);
      consume(j);
    }
    // Tail (<=3 edges): drain and finish.
    __builtin_amdgcn_s_wait_asynccnt(0);
    for (; j < cnt; ++j) consume(j);
#else
    for (int j = 0; j < cnt; ++j) {
      const int cj = __shfl(c, j);
      const float* src = (cj < split) ? (xa + (size_t)cj * 64)
                                      : (xb + (size_t)(cj - split) * 64);
      const float2 xv = *(const float2*)(src + 2 * lane);
      scatter(j, xv.x, xv.y);
    }
#endif
  }
}

// ---------------------------------------------------------------------------
// Launch
// ---------------------------------------------------------------------------
extern "C" void kernel_launch(void* const* d_in, const int* in_sizes, int n_in,
                              void* d_out, int out_size, void* d_ws,
                              size_t ws_size, hipStream_t stream) {
  const float* user_emb  = (const float*)d_in[0];
  const float* item_emb  = (const float*)d_in[1];
  const int*   full_rows = (const int*)d_in[2];
  const int*   full_cols = (const int*)d_in[3];
  const float* full_vals = (const float*)d_in[4];
  const int*   user_rows = (const int*)d_in[5];
  const int*   user_cols = (const int*)d_in[6];
  const float* user_vals = (const float*)d_in[7];
  const int*   item_rows = (const int*)d_in[8];
  const int*   item_cols = (const int*)d_in[9];
  const float* item_vals = (const float*)d_in[10];
  float* out = (float*)d_out;

  const int N_USERS = 100000;
  const int N_ITEMS = 50000;
  const int DIM = 64;

  // 1) zero the whole output buffer
  const int n4 = out_size / 4;
  clagl_zero_f4<<<(n4 + CLAGL_BLOCK - 1) / CLAGL_BLOCK, CLAGL_BLOCK, 0,
                  stream>>>((float4*)out, n4);

  const int nnz_full = in_sizes[2];
  const int nnz_user = in_sizes[5];
  const int nnz_item = in_sizes[8];
  auto nblk = [](int nnz) {
    int b = (nnz + CLAGL_BLOCK - 1) / CLAGL_BLOCK;
    return b < CLAGL_MAXBLK ? b : CLAGL_MAXBLK;
  };

  // 2) full graph: rows 0..149999 land directly on user_emb1 || item_emb1
  clagl_spmm_scatter<<<nblk(nnz_full), CLAGL_BLOCK, 0, stream>>>(
      full_rows, full_cols, full_vals, user_emb, item_emb, N_USERS, out,
      nnz_full);

  // 3) user graph -> user_emb2 at offset (N_USERS+N_ITEMS)*DIM
  clagl_spmm_scatter<<<nblk(nnz_user), CLAGL_BLOCK, 0, stream>>>(
      user_rows, user_cols, user_vals, user_emb, user_emb, 0x7fffffff,
      out + (size_t)(N_USERS + N_ITEMS) * DIM, nnz_user);

  // 4) item graph -> item_emb2 at offset (2*N_USERS+N_ITEMS)*DIM
  clagl_spmm_scatter<<<nblk(nnz_item), CLAGL_BLOCK, 0, stream>>>(
      item_rows, item_cols, item_vals, item_emb, item_emb, 0x7fffffff,
      out + (size_t)(2 * N_USERS + N_ITEMS) * DIM, nnz_item);
}